// HashEmbedModel_40819369181896
// MI455X (gfx1250) — compile-verified
//
#include <hip/hip_runtime.h>
#include <cstdint>
#include <cstddef>

// ---------------------------------------------------------------------------
// HashEmbed model for MI455X (gfx1250, wave32).
//  Stage 1: feature gather (byte_embed + 4 hashed context tables) -> ws
//  Stage 2: GEMM1 [16384x192]x[192x1024] + bias + relu -> hidden (ws)
//  Stage 3: GEMM2 [16384x1024]x[1024x8192] + bias       -> d_out
// GEMMs run on V_WMMA_F32_16X16X4_F32 (exact fp32 semantics) with TDM
// (tensor_load_to_lds) staging A/B tiles into LDS, double buffered,
// using TDM hardware padding for a bank-conflict-free LDS layout.
// ---------------------------------------------------------------------------

#define BM 128
#define BN 64
#define BK 32
#define LDA_T (BK + 4) /* 36 floats per staged A row (TDM pad: 32 DW + 4 DW) */
#define LDB_T (BN + 4) /* 68 floats per staged B row (TDM pad: 64 DW + 4 DW) */

typedef float v2f __attribute__((ext_vector_type(2)));
typedef float v8f __attribute__((ext_vector_type(8)));

// ---------------------------------------------------------------------------
// TDM descriptor-based async tile load (CDNA5 Tensor Data Mover).
// This toolchain (clang-23 / therock-10.0 headers) exposes the 6-arg builtin:
//   (uint32x4 g0, int32x8 g1, int32x4 g2, int32x4 g3, int32x8 g4, i32 cpol)
// Descriptor bitfields per CDNA5 ISA §8.3-8.6; extra group passed as zeros.
// ---------------------------------------------------------------------------
#if defined(__AMDGCN__) && __has_builtin(__builtin_amdgcn_tensor_load_to_lds) && \
    __has_builtin(__builtin_amdgcn_s_wait_tensorcnt)
#define USE_TDM 1
typedef unsigned int v4u_t __attribute__((ext_vector_type(4)));
typedef int v8i_t __attribute__((ext_vector_type(8)));
typedef int v4i_t __attribute__((ext_vector_type(4)));

__device__ __forceinline__ void tdm_load_tile(unsigned lds_byte_addr,
                                              const float* gptr,
                                              unsigned tile_w,      // elements (fast dim)
                                              unsigned tile_h,      // rows
                                              unsigned stride_elems,// row stride in tensor
                                              unsigned pad_interval_code,
                                              unsigned pad_amount_code) {
  const unsigned long long ga = (unsigned long long)(uintptr_t)gptr;
  // --- Group 0: count=1, lds_addr, 57-bit global addr, type=2 ("image")
  v4u_t g0;
  g0[0] = 1u;
  g0[1] = lds_byte_addr;
  g0[2] = (unsigned)ga;
  g0[3] = (unsigned)((ga >> 32) & 0x01FFFFFFu) | 0x80000000u;
  // --- Group 1: wg_mask=0, data_size=4B(2), pad_enable, pad codes,
  //              tensor dims == tile dims (tiles never OOB), dim0 stride.
  v8i_t g1;
  g1[0] = (int)((2u << 16) | (1u << 20) |
                (pad_interval_code << 22) | (pad_amount_code << 25));
  g1[1] = (int)((tile_w & 0xFFFFu) << 16);                               // tensor_dim0[15:0] @48
  g1[2] = (int)(((tile_w >> 16) & 0xFFFFu) | ((tile_h & 0xFFFFu) << 16));// dim0 hi | tensor_dim1 lo
  g1[3] = (int)(((tile_h >> 16) & 0xFFFFu) | ((tile_w & 0xFFFFu) << 16));// dim1 hi | tile_dim0
  g1[4] = (int)(tile_h & 0xFFFFu);                                       // tile_dim1 | tile_dim2=0
  g1[5] = (int)stride_elems;                                             // tensor_dim0_stride lo
  g1[6] = 0;
  g1[7] = 0;
  // --- Group 2/3: higher dims = 1 (unused), strides 0.
  v4i_t g2; g2[0] = 1; g2[1] = 1; g2[2] = 0; g2[3] = 0;
  v4i_t g3; g3[0] = 0; g3[1] = (1 << 16); g3[2] = 0; g3[3] = 0;
  // --- Extra SGPR group required by the 6-arg builtin on this toolchain.
  v8i_t g4; g4[0] = 0; g4[1] = 0; g4[2] = 0; g4[3] = 0;
            g4[4] = 0; g4[5] = 0; g4[6] = 0; g4[7] = 0;
  __builtin_amdgcn_tensor_load_to_lds(g0, g1, g2, g3, g4, 0);
}
#endif

// Flat LDS pointer -> LDS byte offset (ISA: LDS aperture addr[31:0] == offset).
__device__ __forceinline__ unsigned lds_off(const void* p) {
  return (unsigned)(uintptr_t)p;
}

// Manual fallback staging (also keeps host pass valid).
__device__ __forceinline__ void stage_tiles_manual(float* sa, float* sb,
                                                   const float* A, int lda,
                                                   const float* B, int ldb,
                                                   int m0, int n0, int kc, int tid) {
  for (int l = tid; l < (BM * BK) / 4; l += 256) {
    const int row = l >> 3, kq = l & 7;
    const float4 v =
        *(const float4*)(A + (size_t)(m0 + row) * lda + kc * BK + kq * 4);
    *(float4*)&sa[row * LDA_T + kq * 4] = v;
  }
  for (int l = tid; l < (BK * BN) / 4; l += 256) {
    const int row = l >> 4, nq = l & 15;
    const float4 v =
        *(const float4*)(B + (size_t)(kc * BK + row) * ldb + n0 + nq * 4);
    *(float4*)&sb[row * LDB_T + nq * 4] = v;
  }
}

// ---------------------------------------------------------------------------
// Tiled fp32 WMMA GEMM:  C[M x N] = relu?( A[M x K] * B[K x N] + bias )
// Block: 256 threads = 8 wave32; BMxBN = 128x64 tile; wave w owns rows
// [w*16, w*16+16) x all 64 cols as 4 x (16x16) accumulators.
// ---------------------------------------------------------------------------
__global__ __launch_bounds__(256) void gemm_wmma_f32(
    const float* __restrict__ A, int lda,
    const float* __restrict__ B, int ldb,
    const float* __restrict__ bias,
    float* __restrict__ C, int ldc,
    int K, int relu) {
  __shared__ float shA[2][BM * LDA_T];  // 2 x 18 KB
  __shared__ float shB[2][BK * LDB_T];  // 2 x 8.5 KB

  const int m0 = blockIdx.y * BM;
  const int n0 = blockIdx.x * BN;
  const int nk = K / BK;
  const int tid = threadIdx.x;
  const int lane = tid & 31;
  const int wave = tid >> 5;
  const int lm = lane & 15;          // M (A) / N (B) index within fragment
  const int lk2 = (lane >> 4) << 1;  // K base: lanes 16-31 hold K+2,K+3

#if defined(USE_TDM)
  if (wave == 0) {
    tdm_load_tile(lds_off(&shA[0][0]), A + (size_t)m0 * lda,
                  BK, BM, (unsigned)lda, 4u /*32 DW*/, 3u /*+4 DW*/);
    tdm_load_tile(lds_off(&shB[0][0]), B + n0,
                  BN, BK, (unsigned)ldb, 5u /*64 DW*/, 3u /*+4 DW*/);
  }
#else
  stage_tiles_manual(&shA[0][0], &shB[0][0], A, lda, B, ldb, m0, n0, 0, tid);
#endif

  v8f acc[4] = {};

  for (int kc = 0; kc < nk; ++kc) {
    const int cur = kc & 1;
    const int nxt = cur ^ 1;
    // Prefetch next K-chunk into the other buffer while we compute.
    if (kc + 1 < nk) {
#if defined(USE_TDM)
      if (wave == 0) {
        tdm_load_tile(lds_off(&shA[nxt][0]),
                      A + (size_t)m0 * lda + (kc + 1) * BK,
                      BK, BM, (unsigned)lda, 4u, 3u);
        tdm_load_tile(lds_off(&shB[nxt][0]),
                      B + (size_t)(kc + 1) * BK * ldb + n0,
                      BN, BK, (unsigned)ldb, 5u, 3u);
        __builtin_amdgcn_s_wait_tensorcnt(2);  // older (current) pair done
      }
#else
      stage_tiles_manual(&shA[nxt][0], &shB[nxt][0], A, lda, B, ldb, m0, n0,
                         kc + 1, tid);
#endif
    } else {
#if defined(USE_TDM)
      if (wave == 0) __builtin_amdgcn_s_wait_tensorcnt(0);
#endif
    }
    __syncthreads();  // tiles for chunk kc visible to all 8 waves

    const float* As = &shA[cur][(wave * 16 + lm) * LDA_T];
    const float* Bs = &shB[cur][lm];
#pragma unroll
    for (int kk = 0; kk < 8; ++kk) {  // 8 x K=4 steps = BK
      const int k0 = kk * 4 + lk2;
      v2f a;
      a[0] = As[k0];
      a[1] = As[k0 + 1];
#pragma unroll
      for (int n = 0; n < 4; ++n) {
        v2f b;
        b[0] = Bs[k0 * LDB_T + n * 16];
        b[1] = Bs[(k0 + 1) * LDB_T + n * 16];
        acc[n] = __builtin_amdgcn_wmma_f32_16x16x4_f32(
            false, a, false, b, (short)0, acc[n], false, false);
      }
    }
    __syncthreads();  // compute done before buffers are re-filled
  }

  // C/D layout: VGPR r holds M = r + 8*(lane/16); N = lane%16 (+16*n).
  const int row_base = m0 + wave * 16 + (lane >> 4) * 8;
#pragma unroll
  for (int n = 0; n < 4; ++n) {
    const int col = n0 + n * 16 + lm;
    const float bv = bias[col];
#pragma unroll
    for (int r = 0; r < 8; ++r) {
      float v = acc[n][r] + bv;
      if (relu) v = fmaxf(v, 0.0f);
      C[(size_t)(row_base + r) * ldc + col] = v;
    }
  }
}

// ---------------------------------------------------------------------------
// Stage 1: feature gather. One float4 of the 192-float feature row per thread.
// Hash matches reference: h = XOR_k (chars[s-off_k] * prime_k), bucket = h & 0xFFFF,
// with zero contribution when s < off_k (pad-with-zero times prime is 0).
// ---------------------------------------------------------------------------
__device__ __forceinline__ unsigned hash_bucket(const int* __restrict__ row,
                                                int s, int t) {
  unsigned h = 0u;
  switch (t) {
    case 0:  // offsets (1,2), primes[0],[1]
      if (s >= 1) h ^= (unsigned)row[s - 1] * 2654435761u;
      if (s >= 2) h ^= (unsigned)row[s - 2] * 2246822519u;
      break;
    case 1:  // offsets (1,2,3), primes[3],[4],[5]
      if (s >= 1) h ^= (unsigned)row[s - 1] * 2028178513u;
      if (s >= 2) h ^= (unsigned)row[s - 2] * 1220703125u;
      if (s >= 3) h ^= (unsigned)row[s - 3] * 1610612741u;
      break;
    case 2:  // offsets (1,3), primes[6],[7]
      if (s >= 1) h ^= (unsigned)row[s - 1] * 805306457u;
      if (s >= 3) h ^= (unsigned)row[s - 3] * 402653189u;
      break;
    default:  // offsets (2,3), primes[(9)%8]=[1], [(10)%8]=[2]
      if (s >= 2) h ^= (unsigned)row[s - 2] * 2246822519u;
      if (s >= 3) h ^= (unsigned)row[s - 3] * 3266489917u;
      break;
  }
  return h & 65535u;
}

__global__ __launch_bounds__(256) void features_kernel(
    const int* __restrict__ chars,
    const float* __restrict__ byte_embed,
    const float* __restrict__ tables,
    float* __restrict__ features) {
  const int gid = blockIdx.x * 256 + threadIdx.x;  // [0, 16384*48)
  const int tok = gid / 48;
  const int j = gid % 48;  // float4 index within 192-float feature row
  const int s = tok & 1023;
  const int* row = chars + (size_t)(tok >> 10) * 1024;
  const float4* src;
  if (j < 16) {  // byte_embed part: 64 floats = 16 float4
    const int c = row[s];
    src = (const float4*)(byte_embed + (size_t)c * 64) + j;
  } else {  // table t part: 32 floats = 8 float4 each
    const int t = (j - 16) >> 3;
    const int jj = (j - 16) & 7;
    const unsigned idx = hash_bucket(row, s, t);
    src = (const float4*)(tables + ((size_t)t * 65536u + idx) * 32) + jj;
  }
  ((float4*)(features + (size_t)tok * 192))[j] = *src;
}

// ---------------------------------------------------------------------------
// Launch. Inputs (setup_inputs order):
//  0 chars[16,1024] i32 | 1 byte_embed[8192,64] f32 | 2 tables[4,65536,32] f32
//  3 W1[192,1024] | 4 b1[1024] | 5 W2[1024,8192] | 6 b2[8192]
// ws: features (12.6 MB) + hidden (67 MB).
// ---------------------------------------------------------------------------
extern "C" void kernel_launch(void* const* d_in, const int* in_sizes, int n_in,
                              void* d_out, int out_size, void* d_ws,
                              size_t ws_size, hipStream_t stream) {
  (void)in_sizes; (void)n_in; (void)out_size; (void)ws_size;
  const int* chars = (const int*)d_in[0];
  const float* byte_embed = (const float*)d_in[1];
  const float* tables = (const float*)d_in[2];
  const float* W1 = (const float*)d_in[3];
  const float* b1 = (const float*)d_in[4];
  const float* W2 = (const float*)d_in[5];
  const float* b2 = (const float*)d_in[6];

  float* features = (float*)d_ws;                       // [16384 x 192]
  float* hidden = features + (size_t)16384 * 192;       // [16384 x 1024]
  float* out = (float*)d_out;                           // [16384 x 8192]

  // Stage 1: 16384 tokens * 48 float4 = 786432 threads.
  features_kernel<<<(16384 * 48) / 256, 256, 0, stream>>>(chars, byte_embed,
                                                          tables, features);
  // Stage 2: hidden = relu(features @ W1 + b1); K=192 (6 chunks of 32).
  {
    dim3 grid(1024 / BN, 16384 / BM);
    gemm_wmma_f32<<<grid, 256, 0, stream>>>(features, 192, W1, 1024, b1,
                                            hidden, 1024, 192, 1);
  }
  // Stage 3: logits = hidden @ W2 + b2; K=1024 (32 chunks of 32).
  {
    dim3 grid(8192 / BN, 16384 / BM);
    gemm_wmma_f32<<<grid, 256, 0, stream>>>(hidden, 1024, W2, 8192, b2,
                                            out, 8192, 1024, 0);
  }
}